// APPNP_7885559956091
// MI455X (gfx1250) — compile-verified
//
#include <hip/hip_runtime.h>
#include <hip/hip_bf16.h>

#define FEATS   512
#define HIDDEN  64
#define CLASSES 32
#define ALPHA   0.1f
#define DEPTH   10
#define SCAN_T  1024

typedef float v2f __attribute__((ext_vector_type(2)));
typedef float v8f __attribute__((ext_vector_type(8)));

// ---------------------------------------------------------------- utilities
__global__ void k_zero_f(float* __restrict__ p, int n) {
    int i = blockIdx.x * blockDim.x + threadIdx.x;
    if (i < n) p[i] = 0.0f;
}
__global__ void k_zero_i(int* __restrict__ p, int n) {
    int i = blockIdx.x * blockDim.x + threadIdx.x;
    if (i < n) p[i] = 0;
}
__global__ void k_copy_i(const int* __restrict__ a, int* __restrict__ b, int n) {
    int i = blockIdx.x * blockDim.x + threadIdx.x;
    if (i < n) b[i] = a[i];
}

// degI[dst[e]] += 1  (native u32 atomic)
__global__ void k_deg(const int* __restrict__ dst, int* __restrict__ degI, int E) {
    int e = blockIdx.x * blockDim.x + threadIdx.x;
    if (e < E) atomicAdd(&degI[dst[e]], 1);
}

// single-workgroup exclusive scan: offs[0..n] from degI[0..n-1] (offs[n] = E)
__global__ __launch_bounds__(SCAN_T) void k_scan(const int* __restrict__ cnt,
                                                 int* __restrict__ offs, int n) {
    __shared__ int tmp[SCAN_T];
    __shared__ int carry;
    if (threadIdx.x == 0) carry = 0;
    __syncthreads();
    for (int base = 0; base < n; base += SCAN_T) {
        int i = base + threadIdx.x;
        int v = (i < n) ? cnt[i] : 0;
        tmp[threadIdx.x] = v;
        __syncthreads();
        for (int d = 1; d < SCAN_T; d <<= 1) {
            int t = (threadIdx.x >= d) ? tmp[threadIdx.x - d] : 0;
            __syncthreads();
            tmp[threadIdx.x] += t;
            __syncthreads();
        }
        if (i < n) offs[i] = carry + (tmp[threadIdx.x] - v);
        __syncthreads();
        if (threadIdx.x == 0) carry += tmp[SCAN_T - 1];
        __syncthreads();
    }
    if (threadIdx.x == 0) offs[n] = carry;
}

// place edges in dst-sorted (CSR) order; norm computed from int degrees
__global__ void k_fill(const int* __restrict__ src, const int* __restrict__ dst,
                       const int* __restrict__ degI, int* __restrict__ cursor,
                       int* __restrict__ src_s, float* __restrict__ norm_s, int E) {
    int e = blockIdx.x * blockDim.x + threadIdx.x;
    if (e < E) {
        int s = src[e], d = dst[e];
        float ds = fmaxf((float)degI[s], 1.0f);
        float dd = fmaxf((float)degI[d], 1.0f);
        int p = atomicAdd(&cursor[d], 1);
        src_s[p]  = s;
        norm_s[p] = rsqrtf(ds * dd);
    }
}

// norm array for the fallback (scatter) path
__global__ void k_norm(const int* __restrict__ src, const int* __restrict__ dst,
                       const int* __restrict__ degI, float* __restrict__ norm, int E) {
    int e = blockIdx.x * blockDim.x + threadIdx.x;
    if (e < E) {
        float ds = fmaxf((float)degI[src[e]], 1.0f);
        float dd = fmaxf((float)degI[dst[e]], 1.0f);
        norm[e]  = rsqrtf(ds * dd);
    }
}

// ---------------------------------------------------------------- MLP (WMMA f32)
// One block = 16-row tile, 4 waves; wave owns a 16-wide HIDDEN slice; layer2 on
// waves 0..1.  V_WMMA_F32_16X16X4_F32 fragments: lanes 0-15 K=0,1; lanes 16-31 K=2,3.
__global__ __launch_bounds__(128) void k_mlp(
        const float* __restrict__ x,
        const float* __restrict__ W1, const float* __restrict__ b1,
        const float* __restrict__ W2, const float* __restrict__ b2,
        float* __restrict__ h0, float* __restrict__ h, int N) {
    __shared__ float xs[16 * 132];   // stride 132 (%64==4): conflict-free column reads
    __shared__ float hs[16 * 68];

    const int tid   = threadIdx.x;
    const int wave  = tid >> 5;
    const int lane  = tid & 31;
    const int laneM = lane & 15;
    const int hi    = lane >> 4;
    const int koff  = hi * 2;
    const int m0    = blockIdx.x * 16;

    v8f acc = {};
    for (int kb = 0; kb < FEATS; kb += 128) {
        for (int idx = tid; idx < 16 * 128; idx += 128) {
            int r = idx >> 7, c = idx & 127;
            int row = m0 + r; if (row >= N) row = N - 1;
            xs[r * 132 + c] = x[(long long)row * FEATS + kb + c];
        }
        __syncthreads();
        const int n0 = wave * 16;
        for (int k = 0; k < 128; k += 4) {
            v2f a, b;
            a.x = xs[laneM * 132 + k + koff];
            a.y = xs[laneM * 132 + k + koff + 1];
            int kg = kb + k + koff;
            b.x = W1[(kg)     * HIDDEN + n0 + laneM];
            b.y = W1[(kg + 1) * HIDDEN + n0 + laneM];
            acc = __builtin_amdgcn_wmma_f32_16x16x4_f32(
                    false, a, false, b, (short)0, acc, false, false);
        }
        __syncthreads();
    }
    {
        const int n0   = wave * 16;
        const float bv = b1[n0 + laneM];
#pragma unroll
        for (int r = 0; r < 8; ++r)
            hs[(r + hi * 8) * 68 + n0 + laneM] = fmaxf(acc[r] + bv, 0.0f);
    }
    __syncthreads();

    if (wave < 2) {                       // wave-uniform: EXEC stays full for WMMA
        const int n0 = wave * 16;
        v8f acc2 = {};
        for (int k = 0; k < HIDDEN; k += 4) {
            v2f a, b;
            a.x = hs[laneM * 68 + k + koff];
            a.y = hs[laneM * 68 + k + koff + 1];
            b.x = W2[(k + koff)     * CLASSES + n0 + laneM];
            b.y = W2[(k + koff + 1) * CLASSES + n0 + laneM];
            acc2 = __builtin_amdgcn_wmma_f32_16x16x4_f32(
                    false, a, false, b, (short)0, acc2, false, false);
        }
        const float bv = b2[n0 + laneM];
#pragma unroll
        for (int r = 0; r < 8; ++r) {
            int row = m0 + r + hi * 8;
            if (row < N) {
                float v = acc2[r] + bv;
                h0[(long long)row * CLASSES + n0 + laneM] = v;
                h [(long long)row * CLASSES + n0 + laneM] = v;
            }
        }
    }
}

// ---------------------------------------------------------------- pull propagation
// One wave per node, lane = class column (32 classes == wave32). Atomic-free:
// coalesced 128B row gathers of hin[src], broadcast norm, fused alpha-combine.
__global__ __launch_bounds__(256) void k_pull(
        const int* __restrict__ offs, const int* __restrict__ src_s,
        const float* __restrict__ norm_s, const float* __restrict__ hin,
        const float* __restrict__ h0, float* __restrict__ hout, int N) {
    int node = (int)((blockIdx.x * (unsigned)blockDim.x + threadIdx.x) >> 5);
    int lane = threadIdx.x & 31;
    if (node >= N) return;
    int a = offs[node], b = offs[node + 1];
    float acc0 = 0.f, acc1 = 0.f, acc2 = 0.f, acc3 = 0.f;
    int j = a;
    for (; j + 4 <= b; j += 4) {
        int   s0 = src_s[j],  s1 = src_s[j + 1], s2 = src_s[j + 2], s3 = src_s[j + 3];
        float n0 = norm_s[j], n1 = norm_s[j + 1], n2 = norm_s[j + 2], n3 = norm_s[j + 3];
        acc0 += hin[s0 * CLASSES + lane] * n0;
        acc1 += hin[s1 * CLASSES + lane] * n1;
        acc2 += hin[s2 * CLASSES + lane] * n2;
        acc3 += hin[s3 * CLASSES + lane] * n3;
    }
    for (; j < b; ++j)
        acc0 += hin[src_s[j] * CLASSES + lane] * norm_s[j];
    float acc = (acc0 + acc1) + (acc2 + acc3);
    int o = node * CLASSES + lane;
    hout[o] = (1.0f - ALPHA) * acc + ALPHA * h0[o];
}

// ---------------------------------------------------------------- scatter fallback
__global__ void k_scatter(const int* __restrict__ src, const int* __restrict__ dst,
                          const float* __restrict__ norm, const float* __restrict__ h,
                          float* __restrict__ agg, int E) {
    long long t = (long long)blockIdx.x * blockDim.x + threadIdx.x;
    if (t >= (long long)E * 8) return;
    int e = (int)(t >> 3);
    int c = ((int)t & 7) * 4;
    float nm = norm[e];
    const float4 hv = *(const float4*)(h + (long long)src[e] * CLASSES + c);
    float* ap = agg + (long long)dst[e] * CLASSES + c;
    __hip_atomic_fetch_add(ap + 0, hv.x * nm, __ATOMIC_RELAXED, __HIP_MEMORY_SCOPE_AGENT);
    __hip_atomic_fetch_add(ap + 1, hv.y * nm, __ATOMIC_RELAXED, __HIP_MEMORY_SCOPE_AGENT);
    __hip_atomic_fetch_add(ap + 2, hv.z * nm, __ATOMIC_RELAXED, __HIP_MEMORY_SCOPE_AGENT);
    __hip_atomic_fetch_add(ap + 3, hv.w * nm, __ATOMIC_RELAXED, __HIP_MEMORY_SCOPE_AGENT);
}
__global__ void k_combine(float* __restrict__ agg, const float* __restrict__ h0,
                          float* __restrict__ out, int n) {
    int i = blockIdx.x * blockDim.x + threadIdx.x;
    if (i < n) {
        out[i] = (1.0f - ALPHA) * agg[i] + ALPHA * h0[i];
        agg[i] = 0.0f;
    }
}

// ---------------------------------------------------------------- launcher
extern "C" void kernel_launch(void* const* d_in, const int* in_sizes, int n_in,
                              void* d_out, int out_size, void* d_ws, size_t ws_size,
                              hipStream_t stream) {
    const float* x     = (const float*)d_in[0];
    const int*   edges = (const int*)  d_in[1];
    const float* W1    = (const float*)d_in[2];
    const float* b1    = (const float*)d_in[3];
    const float* W2    = (const float*)d_in[4];
    const float* b2    = (const float*)d_in[5];

    const int N  = in_sizes[0] / FEATS;   // 100000
    const int E  = in_sizes[1] / 2;       // 3200000
    const int NC = N * CLASSES;
    const int* src = edges;
    const int* dst = edges + E;

    auto pad64 = [](size_t v) { return (v + 63) & ~(size_t)63; };

    // CSR (pull) layout, in 4-byte words
    size_t oDeg  = 0;
    size_t oOffs = oDeg  + pad64((size_t)N);
    size_t oCur  = oOffs + pad64((size_t)N + 1);
    size_t oSrcS = oCur  + pad64((size_t)N);
    size_t oNrmS = oSrcS + pad64((size_t)E);
    size_t oH0   = oNrmS + pad64((size_t)E);
    size_t oHa   = oH0   + pad64((size_t)NC);
    size_t oHb   = oHa   + pad64((size_t)NC);
    size_t needPull = (oHb + pad64((size_t)NC)) * 4;

    int*   wsi = (int*)d_ws;
    float* wsf = (float*)d_ws;

    if (ws_size >= needPull) {
        // ---------------- CSR pull path (atomic-free propagation) ----------------
        int*   degI   = wsi + oDeg;
        int*   offs   = wsi + oOffs;
        int*   cursor = wsi + oCur;
        int*   src_s  = wsi + oSrcS;
        float* norm_s = wsf + oNrmS;
        float* h0     = wsf + oH0;
        float* ha     = wsf + oHa;
        float* hb     = wsf + oHb;

        k_zero_i<<<(N + 255) / 256, 256, 0, stream>>>(degI, N);
        k_deg   <<<(E + 255) / 256, 256, 0, stream>>>(dst, degI, E);
        k_scan  <<<1, SCAN_T, 0, stream>>>(degI, offs, N);
        k_copy_i<<<(N + 255) / 256, 256, 0, stream>>>(offs, cursor, N);
        k_fill  <<<(E + 255) / 256, 256, 0, stream>>>(src, dst, degI, cursor,
                                                      src_s, norm_s, E);
        k_mlp<<<(N + 15) / 16, 128, 0, stream>>>(x, W1, b1, W2, b2, h0, ha, N);

        float* hin = ha; float* hot = hb;
        const int pblocks = (int)(((size_t)N * 32 + 255) / 256);
        for (int it = 0; it < DEPTH; ++it) {
            float* tgt = (it == DEPTH - 1) ? (float*)d_out : hot;
            k_pull<<<pblocks, 256, 0, stream>>>(offs, src_s, norm_s, hin, h0, tgt, N);
            float* t2 = hin; hin = hot; hot = t2;
        }
    } else {
        // ---------------- atomic scatter fallback --------------------------------
        int*   degI = wsi;                                    // N
        float* norm = wsf + pad64((size_t)N);                 // E
        float* h0   = norm + pad64((size_t)E);                // NC
        float* h    = h0 + pad64((size_t)NC);                 // NC
        float* agg  = h  + pad64((size_t)NC);                 // NC

        k_zero_i<<<(N  + 255) / 256, 256, 0, stream>>>(degI, N);
        k_zero_f<<<(NC + 255) / 256, 256, 0, stream>>>(agg, NC);
        k_deg   <<<(E + 255) / 256, 256, 0, stream>>>(dst, degI, E);
        k_norm  <<<(E + 255) / 256, 256, 0, stream>>>(src, dst, degI, norm, E);
        k_mlp<<<(N + 15) / 16, 128, 0, stream>>>(x, W1, b1, W2, b2, h0, h, N);

        const long long st = (long long)E * 8;
        const int sblocks  = (int)((st + 255) / 256);
        for (int it = 0; it < DEPTH; ++it) {
            k_scatter<<<sblocks, 256, 0, stream>>>(src, dst, norm, h, agg, E);
            float* tgt = (it == DEPTH - 1) ? (float*)d_out : h;
            k_combine<<<(NC + 255) / 256, 256, 0, stream>>>(agg, h0, tgt, NC);
        }
    }
}